// RetinaNetFocalLoss_19593640804541
// MI455X (gfx1250) — compile-verified
//
#include <hip/hip_runtime.h>
#include <stdint.h>

// ---------------------------------------------------------------------------
// RetinaNet focal loss, fused single pass for MI455X (gfx1250, wave32).
//   traffic: 251MB clas_preds + 12.6MB bbox_preds streamed once (NT loads),
//            anchors (786KB) L2-resident  ->  ~11.3us HBM roofline.
//   compute: 62.8M logits * (v_exp_f32 + v_log_f32 + v_rcp_f32 + ~10 VALU)
//            stays under the memory roofline.
// Normalization by npos is deferred to a finalize kernel so matching, box
// loss and focal loss fuse into one streaming kernel (no global sync needed).
// ---------------------------------------------------------------------------

typedef float v2f __attribute__((ext_vector_type(2)));
typedef float v4f __attribute__((ext_vector_type(4)));
typedef float v8f __attribute__((ext_vector_type(8)));

#define TGT 64  // targets per image (fixed by the reference)
#define LOG2E 1.4426950408889634f
#define LN2 0.6931471805599453f

__device__ __forceinline__ float fast_log(float x) {
  return __builtin_amdgcn_logf(x) * LN2;  // v_log_f32
}

// Wave32 sum via V_WMMA_F32_16X16X4_F32 against an all-ones B matrix.
// A layout (32-bit 16x4): lanes 0-15 -> {K0,K1}, lanes 16-31 -> {K2,K3}.
// With A.vgpr0 = v, A.vgpr1 = 0:  D[m][n] = v[m] + v[m+16] for every n.
// D layout: lanes 0-15 hold rows 0-7 in 8 VGPRs, lanes 16-31 rows 8-15,
// so (per-lane sum of D regs) + (xor-16 shuffle) = full wave sum.
__device__ __forceinline__ float wave_sum(float v) {
#if defined(__gfx1250__) && __has_builtin(__builtin_amdgcn_wmma_f32_16x16x4_f32)
  v2f a; a[0] = v; a[1] = 0.0f;
  v2f bm; bm[0] = 1.0f; bm[1] = 1.0f;
  v8f c = {};
  c = __builtin_amdgcn_wmma_f32_16x16x4_f32(false, a, false, bm, (short)0, c,
                                            false, false);
  float s = ((c[0] + c[1]) + (c[2] + c[3])) + ((c[4] + c[5]) + (c[6] + c[7]));
  s += __shfl_xor(s, 16, 32);
  return s;
#else
  for (int off = 16; off; off >>= 1) v += __shfl_xor(v, off, 32);
  return v;
#endif
}

__device__ __forceinline__ float smooth_l1(float d) {
  float ad = fabsf(d);
  return (ad < 1.0f) ? 0.5f * d * d : ad - 0.5f;
}

__global__ void retina_init(int* __restrict__ npos, float* __restrict__ cl,
                            float* __restrict__ bb, int B) {
  int i = threadIdx.x;
  if (i < B) { npos[i] = 0; cl[i] = 0.0f; bb[i] = 0.0f; }
}

__global__ __launch_bounds__(256) void retina_loss_main(
    const float* __restrict__ clas_preds,  // [B,A,C]
    const float* __restrict__ bbox_preds,  // [B,A,4]
    const float* __restrict__ bbox_tgts,   // [B,T,4] tlbr
    const int* __restrict__ clas_tgts,     // [B,T]
    const float* __restrict__ anchors,     // [A,4] cthw
    int A, int C, int* __restrict__ g_npos, float* __restrict__ g_cl,
    float* __restrict__ g_bb) {
  __shared__ float s_tly[TGT], s_tlx[TGT], s_bry[TGT], s_brx[TGT];
  __shared__ float s_cy[TGT], s_cx[TGT], s_sh[TGT], s_sw[TGT], s_area[TGT];
  __shared__ int s_cls[TGT];
  __shared__ float s_wcl[8], s_wbb[8];
  __shared__ int s_wnp[8];

  const int b = blockIdx.y;
  const int tid = threadIdx.x;
  const int a = blockIdx.x * 256 + tid;

  // Stage targets; replicate reference's tlbr -> cthw -> tlbr round trip.
  if (tid < TGT) {
    v4f t = *(const v4f*)(bbox_tgts + ((size_t)b * TGT + tid) * 4);
    float cy = (t[0] + t[2]) * 0.5f, cx = (t[1] + t[3]) * 0.5f;
    float sh = t[2] - t[0], sw = t[3] - t[1];
    s_cy[tid] = cy; s_cx[tid] = cx; s_sh[tid] = sh; s_sw[tid] = sw;
    s_tly[tid] = cy - sh * 0.5f; s_tlx[tid] = cx - sw * 0.5f;
    s_bry[tid] = cy + sh * 0.5f; s_brx[tid] = cx + sw * 0.5f;
    s_area[tid] = sh * sw;
    s_cls[tid] = clas_tgts[b * TGT + tid];
  }
  __syncthreads();

  float cl = 0.0f, bb = 0.0f;
  bool pos = false;

  if (a < A) {
    v4f an = *(const v4f*)(anchors + (size_t)a * 4);  // L2-resident
    float acy = an[0], acx = an[1], ash = an[2], asw = an[3];
    float atly = acy - ash * 0.5f, atlx = acx - asw * 0.5f;
    float abry = acy + ash * 0.5f, abrx = acx + asw * 0.5f;
    float aarea = ash * asw;

    // argmax IoU over 64 LDS-resident targets (first-max tie break like jnp)
    float best = -2.0f;
    int bi = 0;
#pragma unroll 4
    for (int t = 0; t < TGT; ++t) {
      float iy = fmaxf(atly, s_tly[t]);
      float ix = fmaxf(atlx, s_tlx[t]);
      float jy = fminf(abry, s_bry[t]);
      float jx = fminf(abrx, s_brx[t]);
      float inter = fmaxf(jy - iy, 0.0f) * fmaxf(jx - ix, 0.0f);
      float uni = aarea + s_area[t] - inter;
      float iou = inter * __builtin_amdgcn_rcpf(uni + 1e-8f);
      float v = (s_cls[t] > 0) ? iou : -1.0f;
      if (v > best) { best = v; bi = t; }
    }
    int m = (best > 0.5f) ? bi : ((best < 0.4f) ? -1 : -2);
    pos = (m >= 0);

    if (pos) {  // smooth-L1 box regression (rare path)
      v4f bp = __builtin_nontemporal_load(
          (const v4f*)(bbox_preds + ((size_t)b * A + a) * 4));
      float r0 = ((s_cy[m] - acy) / ash) / 0.1f;
      float r1 = ((s_cx[m] - acx) / asw) / 0.1f;
      float r2 = fast_log(s_sh[m] / ash + 1e-8f) / 0.2f;
      float r3 = fast_log(s_sw[m] / asw + 1e-8f) / 0.2f;
      bb = smooth_l1(bp[0] - r0) + smooth_l1(bp[1] - r1) +
           smooth_l1(bp[2] - r2) + smooth_l1(bp[3] - r3);
    }

    if (m >= -1) {  // focal loss over C logits (hot path, NT-streamed)
      int hot = pos ? (s_cls[m] - 1) : -1;
      const v4f* cp = (const v4f*)(clas_preds + ((size_t)b * A + a) * (size_t)C);
      int c4n = C >> 2;
      for (int c4 = 0; c4 < c4n; ++c4) {
        v4f x4 = __builtin_nontemporal_load(cp + c4);
#pragma unroll
        for (int j = 0; j < 4; ++j) {
          float x = x4[j];
          float e = __builtin_amdgcn_exp2f(-fabsf(x) * LOG2E);  // e^{-|x|}
          float r = __builtin_amdgcn_rcpf(1.0f + e);
          float pe = e * r;                    // e^{-|x|}/(1+e^{-|x|})
          float p = (x >= 0.0f) ? r : pe;      // sigmoid(x)
          float q = (x >= 0.0f) ? pe : r;      // 1 - sigmoid(x)
          float l = __builtin_amdgcn_logf(1.0f + e) * LN2;
          float sp_pos = fmaxf(x, 0.0f) + l;   // softplus(x)  = -logsig(-x)
          float sp_neg = sp_pos - x;           // softplus(-x) = -logsig(x)
          int c = (c4 << 2) + j;
          float t_bg = 0.25f * p * p * sp_pos;
          float t_fg = 0.75f * q * q * sp_neg;
          cl += (c == hot) ? t_fg : t_bg;
        }
      }
    }
  }

  // Block reduction: WMMA wave sums + ballot popcount, 3 atomics per block.
  float clw = wave_sum(cl);
  float bbw = wave_sum(bb);
  unsigned long long bal = __ballot(pos);
  int lane = tid & 31, w = tid >> 5;
  if (lane == 0) { s_wcl[w] = clw; s_wbb[w] = bbw; s_wnp[w] = __popcll(bal); }
  __syncthreads();
  if (tid == 0) {
    float tc = 0.0f, tb = 0.0f;
    int tn = 0;
    for (int i = 0; i < 8; ++i) { tc += s_wcl[i]; tb += s_wbb[i]; tn += s_wnp[i]; }
    atomicAdd(&g_cl[b], tc);
    atomicAdd(&g_bb[b], tb);
    atomicAdd(&g_npos[b], tn);
  }
}

__global__ void retina_final(const int* __restrict__ npos,
                             const float* __restrict__ cl,
                             const float* __restrict__ bb,
                             float* __restrict__ out, int B) {
  if (threadIdx.x == 0 && blockIdx.x == 0) {
    float tot = 0.0f;
    for (int i = 0; i < B; ++i) {
      float n = (float)npos[i];
      tot += bb[i] / fmaxf(4.0f * n, 1.0f) + cl[i] / fmaxf(n, 1.0f);
    }
    out[0] = tot / (float)B;
  }
}

extern "C" void kernel_launch(void* const* d_in, const int* in_sizes, int n_in,
                              void* d_out, int out_size, void* d_ws,
                              size_t ws_size, hipStream_t stream) {
  const float* clas_preds = (const float*)d_in[0];
  const float* bbox_preds = (const float*)d_in[1];
  const float* bbox_tgts = (const float*)d_in[2];
  const int* clas_tgts = (const int*)d_in[3];
  const float* anchors = (const float*)d_in[4];

  const int A = in_sizes[4] / 4;        // 49104
  const int B = in_sizes[3] / TGT;      // 16
  const int C = in_sizes[0] / (B * A);  // 80

  int* g_npos = (int*)d_ws;
  float* g_cl = (float*)(g_npos + B);
  float* g_bb = g_cl + B;

  retina_init<<<1, 256, 0, stream>>>(g_npos, g_cl, g_bb, B);
  dim3 grid((A + 255) / 256, B);
  retina_loss_main<<<grid, 256, 0, stream>>>(clas_preds, bbox_preds, bbox_tgts,
                                             clas_tgts, anchors, A, C, g_npos,
                                             g_cl, g_bb);
  retina_final<<<1, 32, 0, stream>>>(g_npos, g_cl, g_bb, (float*)d_out, B);
}